// SelectiveScan_23450521436676
// MI455X (gfx1250) — compile-verified
//
#include <hip/hip_runtime.h>
#include <hip/hip_bf16.h>

// Mamba selective scan, chunked 3-phase parallel scan for MI455X (gfx1250).
// Phase 1: per-chunk local scan (x0=0) -> chunk decay P and local final state Xloc.
// Phase 2: tiny sequential scan over chunk summaries -> per-chunk initial states (in-place in X).
// Phase 3: replay chunks with correct initial state, emit y = sum_n x*C + u*D.

constexpr int BSZ = 2;
constexpr int LEN = 1024;
constexpr int DIN = 1536;
constexpr int DST = 16;
constexpr int CH  = 64;           // chunk length
constexpr int NC  = LEN / CH;     // 16 chunks
constexpr int BLK = 128;          // threads per scan block (4 waves) -> 384 workgroups

#define LOG2E 1.4426950408889634f

#if defined(__AMDGCN__) && __has_builtin(__builtin_amdgcn_global_load_async_to_lds_b128)
#define HAVE_ASYNC_LDS 1
#else
#define HAVE_ASYNC_LDS 0
#endif

typedef int v4i __attribute__((vector_size(4 * sizeof(int))));
typedef __attribute__((address_space(1))) v4i* g_v4i_p;   // global (AS1) b128 pointer
typedef __attribute__((address_space(3))) v4i* l_v4i_p;   // LDS (AS3) b128 pointer

__device__ __forceinline__ float fast_exp2(float x) {
#if defined(__AMDGCN__) && __has_builtin(__builtin_amdgcn_exp2f)
  return __builtin_amdgcn_exp2f(x);
#else
  return exp2f(x);
#endif
}

// copy 16 bytes global -> LDS (async on gfx1250, per-lane addresses, ASYNCcnt-tracked)
__device__ __forceinline__ void copy16_g2lds(const float* __restrict__ g, float* l) {
#if HAVE_ASYNC_LDS
  __builtin_amdgcn_global_load_async_to_lds_b128(
      (g_v4i_p)g, (l_v4i_p)l, /*offset=*/0, /*cpol=*/0);
#else
  *(float4*)l = *(const float4*)g;
#endif
}

__device__ __forceinline__ void wait_async_lds() {
#if HAVE_ASYNC_LDS
#if __has_builtin(__builtin_amdgcn_s_wait_asynccnt)
  __builtin_amdgcn_s_wait_asynccnt(0);
#else
  asm volatile("s_wait_asynccnt 0" ::: "memory");
#endif
#endif
}

// ---------------- Phase 1 / Phase 3 kernel (templated) ----------------
// grid: (DIN/BLK, NC, BSZ), block: BLK. One d-channel per thread, 16 states in VGPRs.
template <bool WRITE_Y>
__global__ __launch_bounds__(BLK) void mamba_scan_chunk(
    const float* __restrict__ ug, const float* __restrict__ dg,
    const float* __restrict__ Ag, const float* __restrict__ Bg,
    const float* __restrict__ Cg, const float* __restrict__ Dg,
    float* __restrict__ P, float* __restrict__ X, float* __restrict__ y) {
  const int tid = threadIdx.x;
  const int d   = blockIdx.x * BLK + tid;   // 0..1535
  const int c   = blockIdx.y;               // chunk
  const int b   = blockIdx.z;               // batch
  const int l0  = c * CH;

  __shared__ alignas(16) float sB[CH * DST];
  __shared__ alignas(16) float sC[CH * DST];

  // Stage B (and C for phase 3) chunk tiles into LDS: 4KB each = 128 lanes x 32B.
  {
    const int toff = tid * 8;  // float offset, 32B per lane (2 async b128s)
    const size_t gb = ((size_t)b * LEN + l0) * DST + toff;
    copy16_g2lds(Bg + gb,     &sB[toff]);
    copy16_g2lds(Bg + gb + 4, &sB[toff + 4]);
    if (WRITE_Y) {
      copy16_g2lds(Cg + gb,     &sC[toff]);
      copy16_g2lds(Cg + gb + 4, &sC[toff + 4]);
    }
    wait_async_lds();
    __syncthreads();
  }

  // Per-channel A, pre-scaled by log2(e) so exp(dv*A) == exp2(dv*a2).
  float a2[DST];
#pragma unroll
  for (int g = 0; g < 4; ++g) {
    float4 av = *(const float4*)(Ag + (size_t)d * DST + 4 * g);
    a2[4 * g + 0] = av.x * LOG2E;
    a2[4 * g + 1] = av.y * LOG2E;
    a2[4 * g + 2] = av.z * LOG2E;
    a2[4 * g + 3] = av.w * LOG2E;
  }

  const size_t sbase = ((size_t)(b * NC + c)) * (DST * DIN) + d;  // ws summary base
  float x[DST];
  if (WRITE_Y) {
#pragma unroll
    for (int n = 0; n < DST; ++n) x[n] = X[sbase + (size_t)n * DIN];  // chunk-initial state
  } else {
#pragma unroll
    for (int n = 0; n < DST; ++n) x[n] = 0.f;
  }
  const float Dd = WRITE_Y ? Dg[d] : 0.f;

  const float4* sB4 = (const float4*)sB;
  const float4* sC4 = (const float4*)sC;
  const size_t base = ((size_t)b * LEN + l0) * DIN + d;

  float sdelta = 0.f;
#pragma unroll 2
  for (int l = 0; l < CH; ++l) {
    const float dv = dg[base + (size_t)l * DIN];
    const float uv = ug[base + (size_t)l * DIN];
    // stream ahead (global_prefetch_b8; speculative, OOB silently dropped)
    __builtin_prefetch(dg + base + (size_t)(l + 16) * DIN, 0, 1);
    __builtin_prefetch(ug + base + (size_t)(l + 16) * DIN, 0, 1);
    sdelta += dv;
    const float du = dv * uv;
    float acc = 0.f;
#pragma unroll
    for (int g = 0; g < 4; ++g) {
      const float4 bb = sB4[l * 4 + g];  // ds_load_b128, broadcast across lanes
      const float e0 = fast_exp2(a2[4 * g + 0] * dv);
      const float e1 = fast_exp2(a2[4 * g + 1] * dv);
      const float e2 = fast_exp2(a2[4 * g + 2] * dv);
      const float e3 = fast_exp2(a2[4 * g + 3] * dv);
      x[4 * g + 0] = fmaf(e0, x[4 * g + 0], du * bb.x);
      x[4 * g + 1] = fmaf(e1, x[4 * g + 1], du * bb.y);
      x[4 * g + 2] = fmaf(e2, x[4 * g + 2], du * bb.z);
      x[4 * g + 3] = fmaf(e3, x[4 * g + 3], du * bb.w);
      if (WRITE_Y) {
        const float4 cc = sC4[l * 4 + g];
        acc = fmaf(x[4 * g + 0], cc.x, acc);
        acc = fmaf(x[4 * g + 1], cc.y, acc);
        acc = fmaf(x[4 * g + 2], cc.z, acc);
        acc = fmaf(x[4 * g + 3], cc.w, acc);
      }
    }
    if (WRITE_Y) y[base + (size_t)l * DIN] = fmaf(uv, Dd, acc);
  }

  if (!WRITE_Y) {
    // chunk summary: P = prod_t exp(dv*A) = exp2(a2 * sum(dv)); Xloc = local final state
#pragma unroll
    for (int n = 0; n < DST; ++n) {
      P[sbase + (size_t)n * DIN] = fast_exp2(a2[n] * sdelta);
      X[sbase + (size_t)n * DIN] = x[n];
    }
  }
}

// ---------------- Phase 2: scan over chunk summaries ----------------
// One thread per (b,n,d); overwrites X[c] with the state *entering* chunk c.
__global__ __launch_bounds__(256) void mamba_chunk_combine(
    const float* __restrict__ P, float* __restrict__ X) {
  const int t   = blockIdx.x * 256 + threadIdx.x;  // (b*DST+n)*DIN + d
  const int b   = t / (DST * DIN);
  const int rem = t - b * (DST * DIN);
  float run = 0.f;
#pragma unroll
  for (int c = 0; c < NC; ++c) {
    const size_t idx = ((size_t)(b * NC + c)) * (DST * DIN) + rem;
    const float p  = P[idx];
    const float xl = X[idx];
    X[idx] = run;                 // state entering chunk c
    run = fmaf(p, run, xl);       // state leaving chunk c
  }
}

extern "C" void kernel_launch(void* const* d_in, const int* in_sizes, int n_in,
                              void* d_out, int out_size, void* d_ws, size_t ws_size,
                              hipStream_t stream) {
  (void)in_sizes; (void)n_in; (void)out_size; (void)ws_size;
  const float* u  = (const float*)d_in[0];
  const float* dl = (const float*)d_in[1];
  const float* A  = (const float*)d_in[2];
  const float* Bm = (const float*)d_in[3];
  const float* Cm = (const float*)d_in[4];
  const float* Dv = (const float*)d_in[5];
  float* y = (float*)d_out;

  const size_t summ = (size_t)BSZ * NC * DST * DIN;  // 786432 floats = 3 MB
  float* P = (float*)d_ws;
  float* X = P + summ;                               // total ws use: 6 MB

  dim3 grid(DIN / BLK, NC, BSZ);
  mamba_scan_chunk<false><<<grid, BLK, 0, stream>>>(u, dl, A, Bm, Cm, Dv, P, X, y);
  mamba_chunk_combine<<<(BSZ * DST * DIN) / 256, 256, 0, stream>>>(P, X);
  mamba_scan_chunk<true><<<grid, BLK, 0, stream>>>(u, dl, A, Bm, Cm, Dv, P, X, y);
}